// InteractionBlock_15161234555433
// MI455X (gfx1250) — compile-verified
//
#include <hip/hip_runtime.h>
#include <hip/hip_bf16.h>

// ---------------------------------------------------------------------------
// CDNA5 (gfx1250) InteractionBlock: f32 WMMA 16x16x4 GEMMs.
// Edge MLP weights staged in LDS in WMMA-fragment order (ds_load_b64 operands,
// conflict-free); scatter-mean via f32 atomics (targets L2-resident).
// ---------------------------------------------------------------------------

typedef float v2f __attribute__((ext_vector_type(2)));
typedef float v8f __attribute__((ext_vector_type(8)));

#define DV   128
#define DE0  64
#define DE1  128
#define WAVES_PER_BLOCK 4
#define SHIFT_LN2 0.6931471805599453f

// softplus(x) - ln(2), numerically safe
__device__ __forceinline__ float sp_act(float v) {
    float s = (v > 20.0f) ? v : log1pf(__expf(v));
    return s - SHIFT_LN2;
}

// ---------------------------------------------------------------------------
// WMMA f32 16x16x4 register layouts (ISA 7.12.2):
//   A (16x4):  lane l holds row m=l&15; VGPR j holds K = 2*(l>>4)+j
//   B (4x16):  lane l holds col n=l&15; VGPR j holds K = 2*(l>>4)+j
//   C/D(16x16):lane l holds col n=l&15; VGPR r holds row m = 8*(l>>4)+r
// ---------------------------------------------------------------------------

// B kept in global memory (row-major [K x 128]); two strided b32 loads / tile.
template <typename PtrT>
__device__ __forceinline__ void wmma_gemm_strip(PtrT Arow, const float* B, int K,
                                                v8f acc[8], int half, int l16,
                                                float ascale) {
    for (int k0 = 0; k0 < K; k0 += 4) {
        v2f a;
        a.x = Arow[k0 + 2 * half + 0] * ascale;
        a.y = Arow[k0 + 2 * half + 1] * ascale;
#pragma unroll
        for (int t = 0; t < 8; ++t) {
            const float* bp = B + (size_t)(k0 + 2 * half) * 128 + t * 16 + l16;
            v2f b;
            b.x = bp[0];
            b.y = bp[128];
            acc[t] = __builtin_amdgcn_wmma_f32_16x16x4_f32(
                false, a, false, b, (short)0, acc[t], false, false);
        }
    }
}

// B pre-staged in LDS in fragment order: Bfrag[((k0>>2)*8 + t)*32 + lane] is
// this lane's v2f operand -> one conflict-free ds_load_b64 per tile per k-step.
template <typename PtrT>
__device__ __forceinline__ void wmma_gemm_ldsb(PtrT Arow, const v2f* Bfrag, int K,
                                               v8f acc[8], int half, int lane,
                                               float ascale) {
    for (int k0 = 0; k0 < K; k0 += 4) {
        v2f a;
        a.x = Arow[k0 + 2 * half + 0] * ascale;
        a.y = Arow[k0 + 2 * half + 1] * ascale;
        const int s8 = (k0 >> 2) << 3;
#pragma unroll
        for (int t = 0; t < 8; ++t) {
            v2f b = Bfrag[(s8 + t) * 32 + lane];
            acc[t] = __builtin_amdgcn_wmma_f32_16x16x4_f32(
                false, a, false, b, (short)0, acc[t], false, false);
        }
    }
}

// stage row-major W[K x 128] -> LDS fragment order (coalesced, one-time)
__device__ __forceinline__ void stage_weight_frag(float* dst, const float* W,
                                                  int K, int tid, int nthreads) {
    for (int i = tid; i < K * 128; i += nthreads) {
        const int k = i >> 7, n = i & 127;
        const int s = k >> 2, j = k & 1, hh = (k >> 1) & 1;
        const int l = hh * 16 + (n & 15), t = n >> 4;
        dst[(((s << 3) + t) * 32 + l) * 2 + j] = W[i];
    }
}

// bias + softplus-shift applied in C/D layout (bias indexed by column only)
__device__ __forceinline__ void bias_softplus(v8f acc[8], const float* bias, int l16) {
#pragma unroll
    for (int t = 0; t < 8; ++t) {
        float bv = bias[t * 16 + l16];
#pragma unroll
        for (int r = 0; r < 8; ++r) acc[t][r] = sp_act(acc[t][r] + bv);
    }
}

// ---------------------------------------------------------------------------
// Kernel 1: x = node_fea @ W_aw1   (no bias, no activation)
// ---------------------------------------------------------------------------
__global__ __launch_bounds__(32 * WAVES_PER_BLOCK)
void k_node_proj(const float* __restrict__ node_fea,
                 const float* __restrict__ W_aw1,
                 float* __restrict__ x, int N) {
    const int lane = threadIdx.x & 31;
    const int wave = threadIdx.x >> 5;
    const int half = lane >> 4, l16 = lane & 15;
    const int row0 = (blockIdx.x * WAVES_PER_BLOCK + wave) * 16;
    if (row0 >= N) return;  // wave-uniform: EXEC stays all-ones for WMMA

    const int arow = min(row0 + l16, N - 1);
    v8f acc[8] = {};
    wmma_gemm_strip(node_fea + (size_t)arow * DV, W_aw1, DV, acc, half, l16, 1.0f);

#pragma unroll
    for (int t = 0; t < 8; ++t)
#pragma unroll
        for (int r = 0; r < 8; ++r) {
            int m = half * 8 + r;
            if (row0 + m < N)
                x[(size_t)(row0 + m) * DV + t * 16 + l16] = acc[t][r];
        }
}

// ---------------------------------------------------------------------------
// Kernel 2: edge MLP (2x GEMM + softplus), gather x[idx2], multiply, atomic
// scatter-add into sums[idx1] and cnt[idx1]. One wave = 16 edges.
// LDS: 32KB W_fg1 frags + 64KB W_fg2 frags + 32KB h-bounce = 128KB / block
// (CDNA5 WGP has 320KB -> 2 blocks resident).
// ---------------------------------------------------------------------------
__global__ __launch_bounds__(32 * WAVES_PER_BLOCK)
void k_edge(const float* __restrict__ edge_fea,
            const int* __restrict__ idx1, const int* __restrict__ idx2,
            const float* __restrict__ W_fg1, const float* __restrict__ b_fg1,
            const float* __restrict__ W_fg2, const float* __restrict__ b_fg2,
            const float* __restrict__ x,
            float* __restrict__ sums, float* __restrict__ cnt, int E) {
    __shared__ float w1t[DE0 * DE1];                  // 32 KB
    __shared__ float w2t[DE1 * DE1];                  // 64 KB
    __shared__ float hbuf[WAVES_PER_BLOCK * 16 * DE1];// 32 KB

    const int lane = threadIdx.x & 31;
    const int wave = threadIdx.x >> 5;
    const int half = lane >> 4, l16 = lane & 15;

    // one-time cooperative staging of both edge weight matrices (frag order)
    stage_weight_frag(w1t, W_fg1, DE0, threadIdx.x, 32 * WAVES_PER_BLOCK);
    stage_weight_frag(w2t, W_fg2, DE1, threadIdx.x, 32 * WAVES_PER_BLOCK);
    __syncthreads();  // before any wave-uniform exit (barrier membership safe)

    const int e0 = (blockIdx.x * WAVES_PER_BLOCK + wave) * 16;
    if (e0 >= E) return;  // wave-uniform

    // ---- GEMM1: h = sp(edge_fea[16 x 64] @ W_fg1[64 x 128] + b_fg1) ----
    const int arow = min(e0 + l16, E - 1);
    v8f acc[8] = {};
    wmma_gemm_ldsb(edge_fea + (size_t)arow * DE0, (const v2f*)w1t, DE0,
                   acc, half, lane, 1.0f);
    bias_softplus(acc, b_fg1, l16);

    // ---- bounce C/D layout -> row-major LDS (per-wave slice, DS in-order) ----
    float* h = hbuf + wave * 16 * DE1;
#pragma unroll
    for (int t = 0; t < 8; ++t)
#pragma unroll
        for (int r = 0; r < 8; ++r)
            h[(half * 8 + r) * DE1 + t * 16 + l16] = acc[t][r];

    // ---- GEMM2: W = sp(h[16 x 128] @ W_fg2[128 x 128] + b_fg2) ----
    v8f acc2[8] = {};
    wmma_gemm_ldsb(h + l16 * DE1, (const v2f*)w2t, DE1, acc2, half, lane, 1.0f);
    bias_softplus(acc2, b_fg2, l16);

    // ---- gather x[idx2], multiply, scatter-add into sums[idx1] ----
#pragma unroll
    for (int r = 0; r < 8; ++r) {
        const int e = e0 + half * 8 + r;
        if (e < E) {
            const int n2 = idx2[e];
            const int n1 = idx1[e];
#pragma unroll
            for (int t = 0; t < 8; ++t) {
                float v = x[(size_t)n2 * DV + t * 16 + l16] * acc2[t][r];
                atomicAdd(&sums[(size_t)n1 * DV + t * 16 + l16], v);
            }
        }
    }
    if (lane < 16 && (e0 + lane) < E) atomicAdd(&cnt[idx1[e0 + lane]], 1.0f);
}

// ---------------------------------------------------------------------------
// Kernel 3: mean = sums / max(cnt,1); h = sp(mean@W_aw2+b2);
//           out = node_fea + h@W_aw3 + b3
// ---------------------------------------------------------------------------
__global__ __launch_bounds__(32 * WAVES_PER_BLOCK)
void k_out(const float* __restrict__ sums, const float* __restrict__ cnt,
           const float* __restrict__ node_fea,
           const float* __restrict__ W_aw2, const float* __restrict__ b_aw2,
           const float* __restrict__ W_aw3, const float* __restrict__ b_aw3,
           float* __restrict__ out, int N) {
    __shared__ float lds[WAVES_PER_BLOCK * 16 * DV];
    const int lane = threadIdx.x & 31;
    const int wave = threadIdx.x >> 5;
    const int half = lane >> 4, l16 = lane & 15;
    const int row0 = (blockIdx.x * WAVES_PER_BLOCK + wave) * 16;
    if (row0 >= N) return;  // wave-uniform

    // per-row mean scale folded into the A load (lane owns exactly one A row)
    const int arow = min(row0 + l16, N - 1);
    const float rscale = 1.0f / fmaxf(cnt[arow], 1.0f);

    v8f acc[8] = {};
    wmma_gemm_strip(sums + (size_t)arow * DV, W_aw2, DV, acc, half, l16, rscale);
    bias_softplus(acc, b_aw2, l16);

    float* h = lds + wave * 16 * DV;
#pragma unroll
    for (int t = 0; t < 8; ++t)
#pragma unroll
        for (int r = 0; r < 8; ++r)
            h[(half * 8 + r) * DV + t * 16 + l16] = acc[t][r];

    v8f acc2[8] = {};
    wmma_gemm_strip(h + l16 * DV, W_aw3, DV, acc2, half, l16, 1.0f);

#pragma unroll
    for (int t = 0; t < 8; ++t) {
        const float bv = b_aw3[t * 16 + l16];
#pragma unroll
        for (int r = 0; r < 8; ++r) {
            const int m = half * 8 + r;
            if (row0 + m < N) {
                const size_t o = (size_t)(row0 + m) * DV + t * 16 + l16;
                out[o] = node_fea[o] + acc2[t][r] + bv;
            }
        }
    }
}

// ---------------------------------------------------------------------------
extern "C" void kernel_launch(void* const* d_in, const int* in_sizes, int n_in,
                              void* d_out, int out_size, void* d_ws, size_t ws_size,
                              hipStream_t stream) {
    const float* node_fea = (const float*)d_in[0];
    const int*   idx1     = (const int*)d_in[1];
    const int*   idx2     = (const int*)d_in[2];
    const float* edge_fea = (const float*)d_in[3];
    const float* W_aw1    = (const float*)d_in[4];
    const float* W_fg1    = (const float*)d_in[5];
    const float* b_fg1    = (const float*)d_in[6];
    const float* W_fg2    = (const float*)d_in[7];
    const float* b_fg2    = (const float*)d_in[8];
    const float* W_aw2    = (const float*)d_in[9];
    const float* b_aw2    = (const float*)d_in[10];
    const float* W_aw3    = (const float*)d_in[11];
    const float* b_aw3    = (const float*)d_in[12];

    const int N = in_sizes[0] / DV;   // 50000
    const int E = in_sizes[1];        // 640000

    // workspace: x[N*128] | sums[N*128] | cnt[N]
    float* x    = (float*)d_ws;
    float* sums = x + (size_t)N * DV;
    float* cnt  = sums + (size_t)N * DV;

    // zero the scatter targets every call (atomics accumulate)
    hipMemsetAsync(sums, 0, ((size_t)N * DV + N) * sizeof(float), stream);

    const int tpb = 32 * WAVES_PER_BLOCK;
    const int ntile_n = (N + 15) / 16;
    const int ntile_e = (E + 15) / 16;
    const int nblk_n = (ntile_n + WAVES_PER_BLOCK - 1) / WAVES_PER_BLOCK;
    const int nblk_e = (ntile_e + WAVES_PER_BLOCK - 1) / WAVES_PER_BLOCK;

    k_node_proj<<<nblk_n, tpb, 0, stream>>>(node_fea, W_aw1, x, N);
    k_edge<<<nblk_e, tpb, 0, stream>>>(edge_fea, idx1, idx2, W_fg1, b_fg1,
                                       W_fg2, b_fg2, x, sums, cnt, E);
    k_out<<<nblk_n, tpb, 0, stream>>>(sums, cnt, node_fea, W_aw2, b_aw2,
                                      W_aw3, b_aw3, (float*)d_out, N);
}